// MeanLoss_6717328851141
// MI455X (gfx1250) — compile-verified
//
#include <hip/hip_runtime.h>
#include <hip/hip_bf16.h>

typedef __attribute__((ext_vector_type(2))) float v2f;
typedef __attribute__((ext_vector_type(8))) float v8f;

#define N_TOT 131072
#define K_CLS 64
#define D_DIM 256
#define FP_EPS 1e-12f
#define MLDS_STRIDE 260   // 260 % 64 = 4 -> 16 lanes at stride-4 banks, conflict-free

__device__ __forceinline__ v8f wmma4(v2f a, v2f b, v8f c) {
    // V_WMMA_F32_16X16X4_F32 : D(16x16 f32) = A(16x4 f32) x B(4x16 f32) + C
    return __builtin_amdgcn_wmma_f32_16x16x4_f32(
        /*neg_a=*/false, a, /*neg_b=*/false, b,
        /*c_mod=*/(short)0, c, /*reuse_a=*/false, /*reuse_b=*/false);
}

// ---------------------------------------------------------------- zero
__global__ __launch_bounds__(256) void zero_ws(float* __restrict__ ws,
                                               float* __restrict__ out0) {
    int i = blockIdx.x * 256 + threadIdx.x;
    if (i < K_CLS * D_DIM + 2 * K_CLS) ws[i] = 0.0f;
    if (i == 0) out0[0] = 0.0f;
}

// ---------------------------------------------------------------- pass 1
// label_sums[k,d] = sum_n A[n,k] * X[n,d]   (A^T X via WMMA), freq[k] = colsum(A)
__global__ __launch_bounds__(256)
void pass1_colsum(const float* __restrict__ X, const float* __restrict__ A,
                  float* __restrict__ sums, float* __restrict__ freq,
                  int rowsPerBlock) {
    const int tid   = threadIdx.x;
    const int lane  = tid & 31;
    const int wave  = tid >> 5;
    const int ktile = wave & 3;      // 4 k-tiles of 16
    const int dgrp  = wave >> 2;     // 2 d-groups of 128
    const int kbase = ktile * 16;
    const int dbase = dgrp * 128;
    const int l15   = lane & 15;
    const int hi    = (lane >= 16) ? 1 : 0;

    v8f acc[8];
    #pragma unroll
    for (int t = 0; t < 8; ++t) { v8f z = {}; acc[t] = z; }

    float fpart = 0.0f;
    const int row0 = blockIdx.x * rowsPerBlock;

    for (int n0 = row0; n0 < row0 + rowsPerBlock; n0 += 4) {
        const int ra = n0 + (hi ? 2 : 0);
        const int rb = ra + 1;
        // A-fragment: M = k (lanes), K = 4 consecutive n-rows
        v2f af;
        af.x = A[(size_t)ra * K_CLS + kbase + l15];
        af.y = A[(size_t)rb * K_CLS + kbase + l15];
        fpart += af.x + af.y;        // freq for free from the same loads

        const float* xra = X + (size_t)ra * D_DIM + dbase + l15;
        const float* xrb = X + (size_t)rb * D_DIM + dbase + l15;
        #pragma unroll
        for (int dt = 0; dt < 8; ++dt) {
            v2f bf;
            bf.x = xra[dt * 16];
            bf.y = xrb[dt * 16];
            acc[dt] = wmma4(af, bf, acc[dt]);
        }
    }

    // freq: lane L and L+16 hold complementary row-halves for the same k
    fpart += __shfl_xor(fpart, 16, 32);
    if (dgrp == 0 && lane < 16)
        unsafeAtomicAdd(&freq[kbase + lane], fpart);

    // flush partial tiles: C layout -> VGPR r: lanes0-15 M=r, lanes16-31 M=r+8
    #pragma unroll
    for (int dt = 0; dt < 8; ++dt) {
        #pragma unroll
        for (int r = 0; r < 8; ++r) {
            const int k = kbase + r + (hi ? 8 : 0);
            const int d = dbase + dt * 16 + l15;
            unsafeAtomicAdd(&sums[k * D_DIM + d], acc[dt][r]);
        }
    }
}

// ---------------------------------------------------------------- means + m^2
__global__ __launch_bounds__(256)
void means_msq(const float* __restrict__ sums, const float* __restrict__ freq,
               const float* __restrict__ past, float* __restrict__ out_means,
               float* __restrict__ msq) {
    __shared__ float red[8];
    const int k = blockIdx.x;
    const int d = threadIdx.x;
    const float f  = freq[k];
    const float bm = sums[k * D_DIM + d] / fmaxf(f, FP_EPS);
    const float pm = past[k * D_DIM + d];
    const float m  = (f > FP_EPS) ? (pm * 0.5f + bm * 0.5f) : pm;
    out_means[k * D_DIM + d] = m;

    float sq = m * m;
    #pragma unroll
    for (int off = 16; off > 0; off >>= 1) sq += __shfl_xor(sq, off, 32);
    if ((threadIdx.x & 31) == 0) red[threadIdx.x >> 5] = sq;
    __syncthreads();
    if (threadIdx.x == 0) {
        float t = 0.0f;
        #pragma unroll
        for (int i = 0; i < 8; ++i) t += red[i];
        msq[k] = t;
    }
}

// ---------------------------------------------------------------- pass 2
// cross = X @ means^T via WMMA, fused loss reduction
__global__ __launch_bounds__(256)
void pass2_loss(const float* __restrict__ X, const float* __restrict__ A,
                const float* __restrict__ S, const float* __restrict__ means,
                const float* __restrict__ msq, float* __restrict__ loss) {
    extern __shared__ float smem[];
    float* mlds    = smem;                          // 64 x MLDS_STRIDE
    float* msq_lds = smem + K_CLS * MLDS_STRIDE;    // 64
    float* xsq     = msq_lds + K_CLS;               // 8 waves x 16 rows

    const int tid  = threadIdx.x;
    const int lane = tid & 31;
    const int wave = tid >> 5;
    const int l15  = lane & 15;
    const int hi   = (lane >= 16) ? 1 : 0;

    // cooperative stage of label_means (64KB) into padded LDS, float4 both ways
    #pragma unroll
    for (int i = 0; i < 16; ++i) {
        const int idx = tid + i * 256;          // 4096 float4 chunks
        const int k   = idx >> 6;
        const int c4  = idx & 63;
        const float4 v = ((const float4*)means)[(size_t)k * 64 + c4];
        *(float4*)&mlds[k * MLDS_STRIDE + c4 * 4] = v;
    }
    if (tid < K_CLS) msq_lds[tid] = msq[tid];
    if (tid < 128)   xsq[tid] = 0.0f;
    __syncthreads();

    // one 16-row n-tile per wave
    const int n0 = blockIdx.x * 128 + wave * 16;

    v8f acc[4];
    #pragma unroll
    for (int t = 0; t < 4; ++t) { v8f z = {}; acc[t] = z; }

    float xp = 0.0f;
    const float* xrow = X + (size_t)(n0 + l15) * D_DIM + (hi ? 2 : 0);

    for (int d0 = 0; d0 < D_DIM; d0 += 4) {
        v2f af;                       // A: M=n row (lane&15), K = d0..d0+3
        af.x = xrow[d0];
        af.y = xrow[d0 + 1];
        xp += af.x * af.x + af.y * af.y;   // x^2 for free
        #pragma unroll
        for (int kt = 0; kt < 4; ++kt) {
            const float* mrow = &mlds[(kt * 16 + l15) * MLDS_STRIDE + d0 + (hi ? 2 : 0)];
            v2f bf;                   // B: K = d rows, N = k cols
            bf.x = mrow[0];
            bf.y = mrow[1];
            acc[kt] = wmma4(af, bf, acc[kt]);
        }
    }

    // combine the two half-row x^2 partials (lane L covers d%4 in {0,1}, L+16 in {2,3})
    unsafeAtomicAdd(&xsq[wave * 16 + l15], xp);   // ds_add_f32
    __syncthreads();

    float c = 0.0f;
    #pragma unroll
    for (int kt = 0; kt < 4; ++kt) {
        #pragma unroll
        for (int r = 0; r < 8; ++r) {
            const int nl = r + (hi ? 8 : 0);
            const int k  = kt * 16 + l15;
            const float cross = acc[kt][r];
            const float dist  = xsq[wave * 16 + nl] + msq_lds[k] - 2.0f * cross;
            const size_t gi = (size_t)(n0 + nl) * K_CLS + k;
            c += A[gi] * S[gi] * dist;
        }
    }
    #pragma unroll
    for (int off = 16; off > 0; off >>= 1) c += __shfl_xor(c, off, 32);
    if (lane == 0)
        unsafeAtomicAdd(loss, c * (1.0f / (float)N_TOT));
}

// ---------------------------------------------------------------- launch
extern "C" void kernel_launch(void* const* d_in, const int* in_sizes, int n_in,
                              void* d_out, int out_size, void* d_ws, size_t ws_size,
                              hipStream_t stream) {
    const float* X = (const float*)d_in[0];   // embedded       [N, D]
    const float* A = (const float*)d_in[1];   // assignments    [N, K]
    const float* S = (const float*)d_in[2];   // soft_assign    [N, K]
    const float* P = (const float*)d_in[3];   // past_means     [K, D]
    float* out = (float*)d_out;               // [0]=loss, [1..]=label_means (K*D)

    float* ws   = (float*)d_ws;
    float* sums = ws;                         // K*D = 16384
    float* freq = ws + K_CLS * D_DIM;         // 64
    float* msq  = freq + K_CLS;               // 64

    zero_ws<<<(K_CLS * D_DIM + 2 * K_CLS + 255) / 256, 256, 0, stream>>>(ws, out);

    const int rowsPerBlock = 512;             // 256 blocks cover N
    pass1_colsum<<<N_TOT / rowsPerBlock, 256, 0, stream>>>(X, A, sums, freq, rowsPerBlock);

    means_msq<<<K_CLS, 256, 0, stream>>>(sums, freq, P, out + 1, msq);

    const size_t smemBytes = (size_t)(K_CLS * MLDS_STRIDE + K_CLS + 128) * sizeof(float);
    pass2_loss<<<N_TOT / 128, 256, smemBytes, stream>>>(X, A, S, out + 1, msq, out);
}